// LightGCNModel_12068858102251
// MI455X (gfx1250) — compile-verified
//
#include <hip/hip_runtime.h>
#include <hip/hip_bf16.h>

#define DIM 64

typedef unsigned int v4u __attribute__((ext_vector_type(4)));
typedef int          v8i __attribute__((ext_vector_type(8)));
typedef int          v4i __attribute__((ext_vector_type(4)));

// ---------------- degree count ----------------
__global__ void k_zero_i32(int* __restrict__ p, int n) {
    int i = blockIdx.x * blockDim.x + threadIdx.x;
    if (i < n) p[i] = 0;
}

__global__ void k_count(const int* __restrict__ col, int* __restrict__ deg, int E) {
    int e = blockIdx.x * blockDim.x + threadIdx.x;
    if (e < E) atomicAdd(&deg[col[e]], 1);
}

// ---------------- 3-phase exclusive scan of deg -> offs ----------------
__global__ void k_scan_a(const int* __restrict__ deg, int* __restrict__ offs,
                         int* __restrict__ partial, float* __restrict__ dinv, int n) {
    __shared__ int sd[1024];
    const int tid = threadIdx.x;
    const int i = blockIdx.x * 1024 + tid;
    int v = (i < n) ? deg[i] : 0;
    sd[tid] = v;
    __syncthreads();
    for (int off = 1; off < 1024; off <<= 1) {
        int t = (tid >= off) ? sd[tid - off] : 0;
        __syncthreads();
        sd[tid] += t;
        __syncthreads();
    }
    if (i < n) {
        offs[i] = sd[tid] - v;
        dinv[i] = (v > 0) ? (1.0f / sqrtf((float)v)) : 0.0f;
    }
    if (tid == 1023) partial[blockIdx.x] = sd[1023];
}

__global__ void k_scan_b(int* __restrict__ partial, int* __restrict__ offs,
                         int nb, int n, int E) {
    __shared__ int sd[1024];
    const int tid = threadIdx.x;
    int v = (tid < nb) ? partial[tid] : 0;
    sd[tid] = v;
    __syncthreads();
    for (int off = 1; off < 1024; off <<= 1) {
        int t = (tid >= off) ? sd[tid - off] : 0;
        __syncthreads();
        sd[tid] += t;
        __syncthreads();
    }
    if (tid < nb) partial[tid] = sd[tid] - v;
    if (tid == 0) offs[n] = E;
}

__global__ void k_scan_c(int* __restrict__ offs, const int* __restrict__ partial,
                         int* __restrict__ cursor, int n) {
    int i = blockIdx.x * blockDim.x + threadIdx.x;
    if (i < n) {
        int o = offs[i] + partial[i >> 10];
        offs[i] = o;
        cursor[i] = o;
    }
}

// ---------------- CSR build ----------------
__global__ void k_build(const int* __restrict__ row, const int* __restrict__ col,
                        const float* __restrict__ dinv, int* __restrict__ cursor,
                        int* __restrict__ srcs, float* __restrict__ wts, int E) {
    int e = blockIdx.x * blockDim.x + threadIdx.x;
    if (e >= E) return;
    int r = row[e];
    int c = col[e];
    float w = dinv[r] * dinv[c];
    int pos = atomicAdd(&cursor[c], 1);
    srcs[pos] = r;
    wts[pos] = w;
}

// ---------------- init: x0 = emb, acc = emb ----------------
__global__ void k_init(const float4* __restrict__ emb, float4* __restrict__ x0,
                       float4* __restrict__ acc, int n4) {
    int i = blockIdx.x * blockDim.x + threadIdx.x;
    if (i < n4) {
        float4 v = emb[i];
        x0[i] = v;
        acc[i] = v;
    }
}

// ---------------- TDM gather-mode row fetch: 8 rows x 256B -> LDS ----------
// D# packing per CDNA5 ISA sect 8.3/8.4 (gather mode,
// 32-bit indices, data_size=4B, row width 64 elements). 6-arg builtin form
// (amdgpu-toolchain/clang-23): groups 0..3 + unused VADDR4 slot + cpol.
__device__ __forceinline__ void tdm_gather8(unsigned int lds_addr,
                                            const float* __restrict__ base,
                                            const int* idx, int cnt, int nrows) {
    unsigned long long ga = (unsigned long long)(uintptr_t)base;
    v4u g0;
    g0.x = 0xC0000001u;                       // count=1 | 32b indices | gather_mode
    g0.y = lds_addr;                          // LDS byte address of tile
    g0.z = (unsigned int)ga;                  // global_addr[31:0]
    g0.w = (unsigned int)(ga >> 32) | 0x80000000u; // addr[56:32] | type=2
    v8i g1;
    g1.s0 = 0x20000;                          // data_size=2 (4 bytes), wg_mask=0
    g1.s1 = (int)(64u << 16);                 // tensor_dim0 = 64 (low 16 bits)
    g1.s2 = (int)(((unsigned)nrows & 0xFFFFu) << 16);      // tensor_dim1 lo
    g1.s3 = (int)((64u << 16) | ((unsigned)nrows >> 16));  // tensor_dim1 hi | tile_dim0=64
    g1.s4 = cnt;                              // tile_dim1 = #valid row indices
    g1.s5 = 64;                               // tensor_dim0_stride = 64 elements
    g1.s6 = 0;
    g1.s7 = 0;
    v4i g2 = { idx[0], idx[1], idx[2], idx[3] };
    v4i g3 = { idx[4], idx[5], idx[6], idx[7] };
    v8i g4 = { 0, 0, 0, 0, 0, 0, 0, 0 };      // unused VADDR4 slot
    __builtin_amdgcn_tensor_load_to_lds(g0, g1, g2, g3, g4, 0);
}

// ---------------- pull-gather layer via TDM double buffering -------------
// One wave32 per target node; per 8-neighbor chunk one tensor_load_to_lds
// gathers the 8 source rows (2KB) into a wave-private LDS buffer while the
// previous chunk is consumed from the other buffer (TENSORcnt pipelined).
__global__ __launch_bounds__(256)
void k_gather(const int* __restrict__ offs, const int* __restrict__ srcs,
              const float* __restrict__ wts, const float* __restrict__ xin,
              float* __restrict__ xout, float* __restrict__ accp,
              float scale, int n) {
    __shared__ __align__(16) unsigned char smem[8][2][2048]; // 8 waves x 2 bufs x 8 rows

    int t = blockIdx.x * blockDim.x + threadIdx.x;
    int node = t >> 5;
    int lane = t & 31;
    int wid = (threadIdx.x >> 5) & 7;
    if (node >= n) return;

    const int s0 = offs[node];
    const int s1 = offs[node + 1];
    const int len = s1 - s0;
    float2 s = {0.0f, 0.0f};

    if (len > 0) {
        const int nchunks = (len + 7) >> 3;
        unsigned int lds0 = (unsigned int)(uintptr_t)&smem[wid][0][0];
        unsigned int lds1 = (unsigned int)(uintptr_t)&smem[wid][1][0];

        // issue chunk 0
        {
            int idx[8];
#pragma unroll
            for (int j = 0; j < 8; ++j) idx[j] = (s0 + j < s1) ? srcs[s0 + j] : 0;
            int cnt = (len < 8) ? len : 8;
            tdm_gather8(lds0, xin, idx, cnt, n);
        }

        for (int c = 0; c < nchunks; ++c) {
            int k0 = s0 + (c << 3);
            int cnt = s1 - k0; if (cnt > 8) cnt = 8;

            if (c + 1 < nchunks) {
                int k1 = k0 + 8;
                int idx[8];
#pragma unroll
                for (int j = 0; j < 8; ++j) idx[j] = (k1 + j < s1) ? srcs[k1 + j] : 0;
                int cnt1 = s1 - k1; if (cnt1 > 8) cnt1 = 8;
                tdm_gather8((c & 1) ? lds0 : lds1, xin, idx, cnt1, n);
                __builtin_amdgcn_s_wait_tensorcnt(1); // chunk c landed, c+1 in flight
            } else {
                __builtin_amdgcn_s_wait_tensorcnt(0); // drain
            }
            __asm__ volatile("" ::: "memory"); // keep LDS reads below the wait

            const float2* lbuf = (const float2*)&smem[wid][c & 1][0];
            for (int j = 0; j < cnt; ++j) {
                float w = wts[k0 + j];
                float2 xv = lbuf[j * 32 + lane]; // row j, dims (2*lane, 2*lane+1)
                s.x = fmaf(w, xv.x, s.x);
                s.y = fmaf(w, xv.y, s.y);
            }
        }
    }

    size_t idx = (size_t)node * 32 + lane;
    float2 a = ((const float2*)accp)[idx];
    a.x = (a.x + s.x) * scale;
    a.y = (a.y + s.y) * scale;
    ((float2*)accp)[idx] = a;
    ((float2*)xout)[idx] = s;
}

extern "C" void kernel_launch(void* const* d_in, const int* in_sizes, int n_in,
                              void* d_out, int out_size, void* d_ws, size_t ws_size,
                              hipStream_t stream) {
    const int* edge_index = (const int*)d_in[0];   // [2, E]: row then col
    const float* emb      = (const float*)d_in[1]; // [N, 64]
    const int E = in_sizes[0] / 2;
    const int N = in_sizes[1] / DIM;
    const int* rowp = edge_index;
    const int* colp = edge_index + E;

    char* ws = (char*)d_ws;
    auto carve = [&](size_t bytes) -> void* {
        void* p = (void*)ws;
        ws += (bytes + 255) & ~(size_t)255;
        return p;
    };
    int*   deg    = (int*)  carve((size_t)N * sizeof(int));
    int*   offs   = (int*)  carve((size_t)(N + 1) * sizeof(int));
    int*   cursor = (int*)  carve((size_t)N * sizeof(int));
    int*   parts  = (int*)  carve((size_t)1024 * sizeof(int));
    int*   srcs   = (int*)  carve((size_t)E * sizeof(int));
    float* wts    = (float*)carve((size_t)E * sizeof(float));
    float* dinv   = (float*)carve((size_t)N * sizeof(float));
    float* x0     = (float*)carve((size_t)N * DIM * sizeof(float));
    float* x1     = (float*)carve((size_t)N * DIM * sizeof(float));
    float* acc    = (float*)d_out;

    const int B = 256;
    const int gN = (N + B - 1) / B;
    const int gE = (E + B - 1) / B;
    const int nb = (N + 1023) / 1024;
    const int g4 = (N * (DIM / 4) + B - 1) / B;
    const int gG = (N * 32 + B - 1) / B;   // one wave32 per node

    k_zero_i32<<<gN, B, 0, stream>>>(deg, N);
    k_count<<<gE, B, 0, stream>>>(colp, deg, E);

    k_scan_a<<<nb, 1024, 0, stream>>>(deg, offs, parts, dinv, N);
    k_scan_b<<<1, 1024, 0, stream>>>(parts, offs, nb, N, E);
    k_scan_c<<<gN, B, 0, stream>>>(offs, parts, cursor, N);

    k_build<<<gE, B, 0, stream>>>(rowp, colp, dinv, cursor, srcs, wts, E);

    k_init<<<g4, B, 0, stream>>>((const float4*)emb, (float4*)x0, (float4*)acc,
                                 N * (DIM / 4));

    k_gather<<<gG, B, 0, stream>>>(offs, srcs, wts, x0, x1, acc, 1.0f, N);
    k_gather<<<gG, B, 0, stream>>>(offs, srcs, wts, x1, x0, acc, 1.0f, N);
    k_gather<<<gG, B, 0, stream>>>(offs, srcs, wts, x0, x1, acc, 0.25f, N);
}